// GNN_21973052686417
// MI455X (gfx1250) — compile-verified
//
#include <hip/hip_runtime.h>
#include <hip/hip_bf16.h>

typedef __attribute__((ext_vector_type(2))) float v2f;
typedef __attribute__((ext_vector_type(8))) float v8f;

#define D_DIM 128

// ---------------------------------------------------------------------------
// GEMM: H[N,D] = X[N,K] @ W[K,D]  via V_WMMA_F32_16X16X4_F32 (exact fp32)
// Block = 256 threads = 8 waves; blockIdx.x = M-tile (16 rows);
// wave w handles N-tile columns [16w, 16w+16). N multiple of 16 (50000=3125*16),
// K multiple of 4 (256/128), D == 128.
// A layout (ISA 7.12.2, 32-bit A 16x4): lane l -> row M=l&15,
//   v0 = K = (l<16 ? 0 : 2), v1 = K+1.
// B layout (4x16, rows striped across lanes): lane l -> col N=l&15,
//   v0 = row K = (l<16 ? 0 : 2), v1 = row K+1.
// C/D layout: lane l -> col N=l&15, VGPR r -> row M = r + (l>=16 ? 8 : 0).
// ---------------------------------------------------------------------------
__global__ __launch_bounds__(256) void gemm_wmma_f32(
    const float* __restrict__ X, const float* __restrict__ W,
    float* __restrict__ H, int K) {
  const int lane  = threadIdx.x & 31;
  const int wave  = threadIdx.x >> 5;     // 0..7 -> N tile
  const int tileM = blockIdx.x * 16;
  const int tileN = wave * 16;
  const int mn    = lane & 15;            // row for A, col for B/C
  const int hi    = lane >> 4;            // 0 or 1

  const float* __restrict__ xrow = X + (size_t)(tileM + mn) * K;

  v8f c = {};
  for (int k0 = 0; k0 < K; k0 += 4) {
    const int ka = k0 + hi * 2;
    v2f a, b;
    a.x = xrow[ka];
    a.y = xrow[ka + 1];
    b.x = W[(size_t)ka * D_DIM + tileN + mn];
    b.y = W[(size_t)(ka + 1) * D_DIM + tileN + mn];
    c = __builtin_amdgcn_wmma_f32_16x16x4_f32(
        /*neg_a=*/false, a, /*neg_b=*/false, b,
        /*c_mod=*/(short)0, c, /*reuse_a=*/false, /*reuse_b=*/false);
  }

  float* __restrict__ hcol = H + tileN + mn;
#pragma unroll
  for (int r = 0; r < 8; ++r) {
    const int mm = r + hi * 8;
    hcol[(size_t)(tileM + mm) * D_DIM] = c[r];
  }
}

// ---------------------------------------------------------------------------
// Per-node attention logits: al_s[n] = <h[n], a_s>, al_d[n] = <h[n], a_d>
// One wave per node, 8 nodes per 256-thread block.
// ---------------------------------------------------------------------------
__global__ __launch_bounds__(256) void node_logits(
    const float* __restrict__ h, const float* __restrict__ a_s,
    const float* __restrict__ a_d, float* __restrict__ al_s,
    float* __restrict__ al_d, int N) {
  const int node = blockIdx.x * 8 + (threadIdx.x >> 5);
  const int lane = threadIdx.x & 31;
  if (node >= N) return;
  const float* __restrict__ hr = h + (size_t)node * D_DIM;
  float ss = 0.f, sd = 0.f;
#pragma unroll
  for (int i = 0; i < 4; ++i) {
    const int d = lane + 32 * i;
    const float v = hr[d];
    ss += v * a_s[d];
    sd += v * a_d[d];
  }
#pragma unroll
  for (int off = 16; off > 0; off >>= 1) {
    ss += __shfl_down(ss, off, 32);
    sd += __shfl_down(sd, off, 32);
  }
  if (lane == 0) { al_s[node] = ss; al_d[node] = sd; }
}

// Monotonic-encoding float atomic max (exact, order-independent).
__device__ __forceinline__ void atomicMaxF32(float* addr, float val) {
  const int iv = __float_as_int(val);
  if (iv >= 0) atomicMax((int*)addr, iv);
  else         atomicMin((unsigned int*)addr, (unsigned int)iv);
}

__global__ void fill_f32(float* __restrict__ p, float v, size_t n) {
  const size_t i = (size_t)blockIdx.x * 256 + threadIdx.x;
  if (i < n) p[i] = v;
}

// Pass 1: edge logit (leaky-relu) -> elog, segment max over dst.
// Edges e < E come from edge_index; e in [E, E+N) are self loops.
__global__ __launch_bounds__(256) void edge_logit_max(
    const int* __restrict__ src, const int* __restrict__ dst, int E, int N,
    const float* __restrict__ al_s, const float* __restrict__ al_d,
    float* __restrict__ elog, float* __restrict__ m) {
  const int e = blockIdx.x * 256 + threadIdx.x;
  if (e >= E + N) return;
  int s, d;
  if (e < E) { s = src[e]; d = dst[e]; } else { s = d = e - E; }
  float x = al_s[s] + al_d[d];
  x = (x > 0.f) ? x : 0.2f * x;  // NEG_SLOPE = 0.2
  elog[e] = x;
  atomicMaxF32(m + d, x);
}

// Pass 2: p = exp(logit - m[dst]); elog <- p; z[dst] += p.
__global__ __launch_bounds__(256) void edge_exp_sum(
    const int* __restrict__ dst, int E, int N,
    float* __restrict__ elog, const float* __restrict__ m,
    float* __restrict__ z) {
  const int e = blockIdx.x * 256 + threadIdx.x;
  if (e >= E + N) return;
  const int d = (e < E) ? dst[e] : (e - E);
  const float p = expf(elog[e] - m[d]);
  elog[e] = p;
  atomicAdd(z + d, p);
}

// Pass 3: out[dst] += (p / z[dst]) * h[src].  One wave per edge; 128 dims
// => 4 coalesced atomicAdds per lane. h (25.6 MB) is L2-resident.
__global__ __launch_bounds__(256) void edge_scatter(
    const int* __restrict__ src, const int* __restrict__ dst, int E, int N,
    const float* __restrict__ elog, const float* __restrict__ z,
    const float* __restrict__ h, float* __restrict__ out) {
  const int e    = blockIdx.x * 8 + (threadIdx.x >> 5);
  const int lane = threadIdx.x & 31;
  if (e >= E + N) return;
  int s, d;
  if (e < E) { s = src[e]; d = dst[e]; } else { s = d = e - E; }
  const float coef = elog[e] / z[d];
  const float* __restrict__ hs = h + (size_t)s * D_DIM;
  float* __restrict__ od = out + (size_t)d * D_DIM;
#pragma unroll
  for (int i = 0; i < 4; ++i) {
    const int dim = lane + 32 * i;
    atomicAdd(od + dim, coef * hs[dim]);
  }
}

// out += bias; optional ReLU.
__global__ void bias_act(float* __restrict__ h, const float* __restrict__ b,
                         size_t n, int relu) {
  const size_t i = (size_t)blockIdx.x * 256 + threadIdx.x;
  if (i >= n) return;
  float v = h[i] + b[i & (D_DIM - 1)];
  if (relu) v = fmaxf(v, 0.f);
  h[i] = v;
}

// Global mean pool: sums + counts via atomics, then divide.
__global__ __launch_bounds__(256) void pool_sum(
    const float* __restrict__ h, const int* __restrict__ batch,
    float* __restrict__ out, float* __restrict__ cnt, int N) {
  const int node = blockIdx.x * 8 + (threadIdx.x >> 5);
  const int lane = threadIdx.x & 31;
  if (node >= N) return;
  const int g = batch[node];
  const float* __restrict__ hr = h + (size_t)node * D_DIM;
#pragma unroll
  for (int i = 0; i < 4; ++i) {
    const int d = lane + 32 * i;
    atomicAdd(out + (size_t)g * D_DIM + d, hr[d]);
  }
  if (lane == 0) atomicAdd(cnt + g, 1.0f);
}

__global__ void pool_div(float* __restrict__ out, const float* __restrict__ cnt,
                         int total) {
  const int i = blockIdx.x * 256 + threadIdx.x;
  if (i < total) out[i] /= fmaxf(cnt[i >> 7], 1.0f);
}

// ---------------------------------------------------------------------------
// Host-side driver
// ---------------------------------------------------------------------------
static inline int cdiv(int a, int b) { return (a + b - 1) / b; }
static inline int cdivz(size_t a, int b) { return (int)((a + b - 1) / b); }

static void run_gat_layer(hipStream_t stream,
                          const float* xin, int K,
                          const float* W, const float* as_, const float* ad_,
                          const float* bias, int relu,
                          float* hlin, float* hout,
                          float* al_s, float* al_d, float* mbuf, float* zbuf,
                          float* elog,
                          const int* src, const int* dst, int E, int N) {
  const size_t N128 = (size_t)N * D_DIM;
  const int ET = E + N;

  // 1) h = xin @ W (WMMA fp32)
  gemm_wmma_f32<<<N / 16, 256, 0, stream>>>(xin, W, hlin, K);
  // 2) per-node logits
  node_logits<<<cdiv(N, 8), 256, 0, stream>>>(hlin, as_, ad_, al_s, al_d, N);
  // 3) init segment max / sum
  fill_f32<<<cdiv(N, 256), 256, 0, stream>>>(mbuf, -__builtin_inff(), (size_t)N);
  fill_f32<<<cdiv(N, 256), 256, 0, stream>>>(zbuf, 0.f, (size_t)N);
  // 4) edge softmax
  edge_logit_max<<<cdiv(ET, 256), 256, 0, stream>>>(src, dst, E, N, al_s, al_d,
                                                    elog, mbuf);
  edge_exp_sum<<<cdiv(ET, 256), 256, 0, stream>>>(dst, E, N, elog, mbuf, zbuf);
  // 5) aggregate into hout (zeroed first)
  fill_f32<<<cdivz(N128, 256), 256, 0, stream>>>(hout, 0.f, N128);
  edge_scatter<<<cdiv(ET, 8), 256, 0, stream>>>(src, dst, E, N, elog, zbuf,
                                                hlin, hout);
  // 6) bias (+ relu)
  bias_act<<<cdivz(N128, 256), 256, 0, stream>>>(hout, bias, N128, relu);
}

extern "C" void kernel_launch(void* const* d_in, const int* in_sizes, int n_in,
                              void* d_out, int out_size, void* d_ws,
                              size_t ws_size, hipStream_t stream) {
  const float* x     = (const float*)d_in[0];
  const int*   ei    = (const int*)d_in[1];
  const int*   batch = (const int*)d_in[2];
  const float* W0  = (const float*)d_in[3];
  const float* as0 = (const float*)d_in[4];
  const float* ad0 = (const float*)d_in[5];
  const float* b0  = (const float*)d_in[6];
  const float* W1  = (const float*)d_in[7];
  const float* as1 = (const float*)d_in[8];
  const float* ad1 = (const float*)d_in[9];
  const float* b1  = (const float*)d_in[10];
  const float* W2  = (const float*)d_in[11];
  const float* as2 = (const float*)d_in[12];
  const float* ad2 = (const float*)d_in[13];
  const float* b2  = (const float*)d_in[14];

  const int E = in_sizes[1] / 2;        // 800000
  const int N = in_sizes[2];            // 50000
  const int in_dim = in_sizes[0] / N;   // 256
  const int* src = ei;                  // edge_index row 0
  const int* dst = ei + E;              // edge_index row 1

  const size_t N128 = (size_t)N * D_DIM;
  float* ws   = (float*)d_ws;
  float* hA   = ws;                     // N*128  (linear-projection output)
  float* hB   = ws + N128;              // N*128  (aggregation output / next input)
  float* al_s = ws + 2 * N128;          // N
  float* al_d = al_s + N;               // N
  float* mbuf = al_d + N;               // N
  float* zbuf = mbuf + N;               // N
  float* elog = zbuf + N;               // E+N
  float* cnt  = elog + (size_t)(E + N); // 16

  // Layer 0: x[N,256] -> hB (ReLU)
  run_gat_layer(stream, x, in_dim, W0, as0, ad0, b0, /*relu=*/1,
                hA, hB, al_s, al_d, mbuf, zbuf, elog, src, dst, E, N);
  // Layer 1: hB[N,128] -> hB (ReLU); lin goes through hA, hB re-zeroed after use
  run_gat_layer(stream, hB, D_DIM, W1, as1, ad1, b1, /*relu=*/1,
                hA, hB, al_s, al_d, mbuf, zbuf, elog, src, dst, E, N);
  // Layer 2: hB[N,128] -> hB (no ReLU)
  run_gat_layer(stream, hB, D_DIM, W2, as2, ad2, b2, /*relu=*/0,
                hA, hB, al_s, al_d, mbuf, zbuf, elog, src, dst, E, N);

  // global_mean_pool -> d_out [16,128]
  float* out = (float*)d_out;
  fill_f32<<<cdiv(out_size, 256), 256, 0, stream>>>(out, 0.f, (size_t)out_size);
  fill_f32<<<1, 256, 0, stream>>>(cnt, 0.f, (size_t)16);
  pool_sum<<<cdiv(N, 8), 256, 0, stream>>>(hB, batch, out, cnt, N);
  pool_div<<<cdiv(out_size, 256), 256, 0, stream>>>(out, cnt, out_size);
}